// MoE_69123203661943
// MI455X (gfx1250) — compile-verified
//
#include <hip/hip_runtime.h>
#include <stdint.h>

// ---------------------------------------------------------------------------
// MoE forward for MI455X (gfx1250): sparse top-2 routed experts + shared
// expert, fp32 WMMA (V_WMMA_F32_16X16X4_F32). Memory-bound on fp32 weight
// streaming (~570MB @ 23.3TB/s ~ 24.5us floor): keep fp32 end-to-end, cut
// compute 7x vs dense via per-expert token gathering, and stream all weight
// tiles through LDS with double-buffered ASYNC global->LDS copies (ASYNCcnt).
// ---------------------------------------------------------------------------

typedef float v2f __attribute__((ext_vector_type(2)));
typedef float v8f __attribute__((ext_vector_type(8)));
typedef int   v4i_ __attribute__((vector_size(16)));   // async-LDS builtin pointee

#define HID    2048
#define INTER  1408
#define SINTER 2816
#define NEXP   14

#define WAVES  16
#define BLOCK  (WAVES * 32)
#define FC     512                  // inter-dim chunk staged in LDS

// LDS layout (floats)
#define XS_STRIDE (HID + 4)         // x tile rows, padded vs bank conflicts
#define HS_STRIDE (FC + 2)          // h chunk rows
#define KC1   32                    // phase-1 K chunk (per async stage)
#define BST1  (KC1 + 4)             // 36: staged B row stride, phase 1
#define KC2   64                    // phase-2 K chunk
#define BST2  (KC2 + 4)             // 68: staged B row stride, phase 2
#define BW_FLOATS (2 * 2 * 16 * BST1)   // 2304 floats per-wave staging region
#define XS_OFF 0
#define HS_OFF (16 * XS_STRIDE)                  // 32832
#define B_OFF  (HS_OFF + 16 * HS_STRIDE)         // 41056
#define LDS_FLOATS (B_OFF + WAVES * BW_FLOATS)   // 77920 -> 311680 bytes

__device__ __forceinline__ float sigmoidf_(float v) {
    return 1.0f / (1.0f + __expf(-v));
}

// ---- CDNA5 async global->LDS copy (16B per lane), with fallbacks ----------
__device__ __forceinline__ void async_cp16(float* ldst, const float* gsrc) {
#if __has_builtin(__builtin_amdgcn_global_load_async_to_lds_b128)
    __builtin_amdgcn_global_load_async_to_lds_b128(
        (__attribute__((address_space(1))) v4i_*)(uintptr_t)gsrc,
        (__attribute__((address_space(3))) v4i_*)(unsigned)(uintptr_t)ldst,
        0, 0);
#elif defined(__gfx1250__)
    asm volatile("global_load_async_to_lds_b128 %0, %1, off"
                 :
                 : "v"((unsigned)(uintptr_t)ldst),
                   "v"((unsigned long long)(uintptr_t)gsrc)
                 : "memory");
#else
    *(float4*)ldst = *(const float4*)gsrc;
#endif
}

template <int N>
__device__ __forceinline__ void wait_async() {
#if __has_builtin(__builtin_amdgcn_s_wait_asynccnt)
    __builtin_amdgcn_s_wait_asynccnt((unsigned short)N);
#elif defined(__gfx1250__)
    asm volatile("s_wait_asynccnt %0" ::"i"(N));
#endif
    asm volatile("" ::: "memory");   // keep LDS reads after the wait
}

// ---------------------------------------------------------------------------
// Router: one wave32 per token. logits = clsdot * silu(gatedot),
// scores = softmax(|logits|); top-2 over (scores + bias); combine weight
// = 1 + score * extra_scale. Appends (token*2+slot, weight) to per-expert
// lists via atomic counters (list order is irrelevant to the result, so the
// final output stays bitwise deterministic).
// ---------------------------------------------------------------------------
__global__ void router_kernel(const float* __restrict__ x,
                              const float* __restrict__ gate_w,
                              const float* __restrict__ cls_w,
                              const float* __restrict__ extra_scale,
                              const float* __restrict__ extra_bias,
                              int*   __restrict__ cnt,
                              int*   __restrict__ listIdx,
                              float* __restrict__ listW,
                              int T)
{
    int tok  = (int)((blockIdx.x * (unsigned)blockDim.x + threadIdx.x) >> 5);
    int lane = threadIdx.x & 31;
    if (tok >= T) return;

    const float* xr = x + (size_t)tok * HID;

    float score[NEXP];
    #pragma unroll 1
    for (int e = 0; e < NEXP; ++e) {
        const float* cw = cls_w  + (size_t)e * HID;
        const float* gw = gate_w + (size_t)e * HID;
        float dc = 0.0f, dg = 0.0f;
        for (int h = lane; h < HID; h += 32) {
            float xv = xr[h];
            dc = fmaf(xv, cw[h], dc);
            dg = fmaf(xv, gw[h], dg);
        }
        #pragma unroll
        for (int off = 16; off; off >>= 1) {
            dc += __shfl_xor(dc, off, 32);
            dg += __shfl_xor(dg, off, 32);
        }
        score[e] = dc * (dg * sigmoidf_(dg));      // logits
    }

    // softmax over |logits|
    float mx = -1e30f;
    #pragma unroll
    for (int e = 0; e < NEXP; ++e) { score[e] = fabsf(score[e]); mx = fmaxf(mx, score[e]); }
    float sum = 0.0f;
    #pragma unroll
    for (int e = 0; e < NEXP; ++e) { score[e] = __expf(score[e] - mx); sum += score[e]; }
    float inv = 1.0f / sum;
    #pragma unroll
    for (int e = 0; e < NEXP; ++e) score[e] *= inv;

    // top-2 on biased scores (first index wins ties, like lax.top_k)
    float b[NEXP];
    #pragma unroll
    for (int e = 0; e < NEXP; ++e) b[e] = score[e] + extra_bias[e];
    int i1 = 0; float m1 = b[0];
    #pragma unroll
    for (int e = 1; e < NEXP; ++e) if (b[e] > m1) { m1 = b[e]; i1 = e; }
    int i2 = (i1 == 0) ? 1 : 0; float m2 = b[i2];
    #pragma unroll
    for (int e = 0; e < NEXP; ++e) if (e != i1 && b[e] > m2) { m2 = b[e]; i2 = e; }

    if (lane == 0) {
        float w0 = 1.0f + score[i1] * extra_scale[i1];
        float w1 = 1.0f + score[i2] * extra_scale[i2];
        int p0 = atomicAdd(&cnt[i1], 1);
        listIdx[i1 * T + p0] = tok * 2 + 0;
        listW  [i1 * T + p0] = w0;
        int p1 = atomicAdd(&cnt[i2], 1);
        listIdx[i2 * T + p1] = tok * 2 + 1;
        listW  [i2 * T + p1] = w1;
    }
}

// ---------------------------------------------------------------------------
// Fused SwiGLU MLP over one 16-token tile:
//   h = silu(x @ Wg^T) * (x @ Wu^T)    [16 x F], chunked by FC through LDS
//   o = h @ Wd^T                       [16 x HID], accumulated in registers
// Weight tiles are streamed into per-wave LDS staging buffers with
// double-buffered async global->LDS copies; WMMA reads A and B from LDS.
// ---------------------------------------------------------------------------
template <bool SHARED>
__global__ void mlp_tile_kernel(const float* __restrict__ x,
                                const float* __restrict__ Wg,   // [E][F][HID] or [F][HID]
                                const float* __restrict__ Wu,
                                const float* __restrict__ Wd,   // [E][HID][F] or [HID][F]
                                const int*   __restrict__ cnt,
                                const int*   __restrict__ listIdx,
                                const float* __restrict__ listW,
                                float* __restrict__ out,
                                int T, int F)
{
    extern __shared__ float lds[];
    float* xs = lds + XS_OFF;              // [16][XS_STRIDE]
    float* hs = lds + HS_OFF;              // [16][HS_STRIDE]
    __shared__ int   s_tok[16];
    __shared__ int   s_out[16];
    __shared__ float s_w[16];

    int tileBase, rows;
    if constexpr (SHARED) {
        tileBase = blockIdx.x * 16;
        if (tileBase >= T) return;
        rows = min(16, T - tileBase);
    } else {
        int expert = blockIdx.y;
        int c = cnt[expert];
        tileBase = blockIdx.x * 16;
        if (tileBase >= c) return;
        rows = min(16, c - tileBase);
        Wg += (size_t)expert * F * HID;
        Wu += (size_t)expert * F * HID;
        Wd += (size_t)expert * HID * F;
        listIdx += (size_t)expert * T;
        listW   += (size_t)expert * T;
    }

    const int tid = threadIdx.x;
    if (tid < 16) {
        if (tid < rows) {
            if constexpr (SHARED) {
                s_tok[tid] = tileBase + tid;
                s_out[tid] = tileBase + tid;
                s_w[tid]   = 1.0f;
            } else {
                int entry = listIdx[tileBase + tid];
                s_tok[tid] = entry >> 1;
                s_out[tid] = entry;                // row in (token,slot) scratch
                s_w[tid]   = listW[tileBase + tid];
            }
        } else {
            s_tok[tid] = 0; s_out[tid] = -1; s_w[tid] = 0.0f;
        }
    }
    __syncthreads();

    // Stage gathered 16 x HID x-tile into LDS (async, 16B units, coalesced).
    {
        for (int idx = tid; idx < 16 * (HID / 4); idx += BLOCK) {
            int r   = idx >> 9;                 // HID/4 = 512 units per row
            int seg = idx & 511;
            async_cp16(xs + r * XS_STRIDE + seg * 4,
                       x + (size_t)s_tok[r] * HID + seg * 4);
        }
    }
    wait_async<0>();
    __syncthreads();

    const int wave = tid >> 5;
    const int lane = tid & 31;
    const int hi   = lane >> 4;     // 0: lanes 0-15, 1: lanes 16-31
    const int lo   = lane & 15;
    const int koff = hi * 2;        // K offset of the upper half-wave fragment
    float* ldsB = lds + B_OFF + wave * BW_FLOATS;   // per-wave staging region

    // Down-projection accumulators: HID/16 = 128 n-tiles over 16 waves.
    v8f oacc[8];
    #pragma unroll
    for (int j = 0; j < 8; ++j) oacc[j] = {};

    for (int fc = 0; fc < F; fc += FC) {
        const int fw = min(FC, F - fc);           // multiple of KC2 here
        __syncthreads();                          // hs safe to overwrite

        // ---- phase 1: h[:, fc:fc+fw] = silu(x Wg^T) * (x Wu^T) ----
        const int ntiles = fw >> 4;
        for (int nt = wave; nt < ntiles; nt += WAVES) {
            const int nbase = fc + nt * 16;
            v8f gacc = {}, uacc = {};

            // double-buffered async staging of 16x32 tiles of Wg and Wu
            auto stage1 = [&](int b, int kc) {
                #pragma unroll
                for (int i = 0; i < 8; ++i) {
                    int u   = i * 32 + lane;     // 256 16B-units: 2 mats x 16 rows x 8 segs
                    int mat = u >> 7;
                    int row = (u >> 3) & 15;
                    int seg = u & 7;
                    const float* src = (mat ? Wu : Wg) +
                        (size_t)(nbase + row) * HID + kc + seg * 4;
                    async_cp16(ldsB + (b * 2 + mat) * (16 * BST1) + row * BST1 + seg * 4,
                               src);
                }
            };

            int b = 0;
            stage1(0, 0);
            const int nch = HID / KC1;            // 64 chunks
            for (int c = 0; c < nch; ++c) {
                if (c + 1 < nch) { stage1(b ^ 1, (c + 1) * KC1); wait_async<8>(); }
                else             { wait_async<0>(); }
                const float* xa = xs + lo * XS_STRIDE + c * KC1 + koff;
                const float* bg = ldsB + (b * 2 + 0) * (16 * BST1) + lo * BST1 + koff;
                const float* bu = ldsB + (b * 2 + 1) * (16 * BST1) + lo * BST1 + koff;
                #pragma unroll
                for (int kk = 0; kk < KC1; kk += 4) {
                    v2f a  = *(const v2f*)(xa + kk);
                    v2f vg = *(const v2f*)(bg + kk);
                    v2f vu = *(const v2f*)(bu + kk);
                    gacc = __builtin_amdgcn_wmma_f32_16x16x4_f32(
                        false, a, false, vg, (short)0, gacc, false, false);
                    uacc = __builtin_amdgcn_wmma_f32_16x16x4_f32(
                        false, a, false, vu, (short)0, uacc, false, false);
                }
                b ^= 1;
            }
            // D fragment: VGPR r, lanes0-15 -> M=r ; lanes16-31 -> M=r+8; N=lo
            #pragma unroll
            for (int r = 0; r < 8; ++r) {
                const int m = r + 8 * hi;
                float g = gacc[r];
                hs[m * HS_STRIDE + nt * 16 + lo] = (g * sigmoidf_(g)) * uacc[r];
            }
        }
        __syncthreads();

        // ---- phase 2: oacc += h_chunk @ Wd[:, fc:fc+fw]^T ----
        #pragma unroll 1
        for (int j = 0; j < 8; ++j) {
            const int nb = (wave + j * WAVES) * 16;          // output-dim base

            auto stage2 = [&](int b, int kc) {
                #pragma unroll
                for (int i = 0; i < 8; ++i) {
                    int u   = i * 32 + lane;     // 256 units: 16 rows x 16 segs
                    int row = u >> 4;
                    int seg = u & 15;
                    async_cp16(ldsB + b * (16 * BST2) + row * BST2 + seg * 4,
                               Wd + (size_t)(nb + row) * F + fc + kc + seg * 4);
                }
            };

            v8f acc = oacc[j];
            int b = 0;
            stage2(0, 0);
            const int nch = fw / KC2;
            for (int c = 0; c < nch; ++c) {
                if (c + 1 < nch) { stage2(b ^ 1, (c + 1) * KC2); wait_async<8>(); }
                else             { wait_async<0>(); }
                const float* ha = hs + lo * HS_STRIDE + c * KC2 + koff;
                const float* bd = ldsB + b * (16 * BST2) + lo * BST2 + koff;
                #pragma unroll
                for (int kk = 0; kk < KC2; kk += 4) {
                    v2f a  = *(const v2f*)(ha + kk);
                    v2f vb = *(const v2f*)(bd + kk);
                    acc = __builtin_amdgcn_wmma_f32_16x16x4_f32(
                        false, a, false, vb, (short)0, acc, false, false);
                }
                b ^= 1;
            }
            oacc[j] = acc;
        }
    }
    __syncthreads();

    // Epilogue: scale rows by combine weight, scatter to output rows.
    #pragma unroll 1
    for (int j = 0; j < 8; ++j) {
        const int nb = (wave + j * WAVES) * 16;
        #pragma unroll
        for (int r = 0; r < 8; ++r) {
            const int m = r + 8 * hi;
            const int orow = s_out[m];
            if (orow >= 0)
                out[(size_t)orow * HID + nb + lo] = oacc[j][r] * s_w[m];
        }
    }
}

// ---------------------------------------------------------------------------
// out[t] += y_slot0[t] + y_slot1[t]   (shared-expert z already in out)
// ---------------------------------------------------------------------------
__global__ void combine_kernel(float* __restrict__ out,
                               const float* __restrict__ ybuf,
                               size_t n /* T*HID */)
{
    size_t i = ((size_t)blockIdx.x * blockDim.x + threadIdx.x) * 4;
    if (i >= n) return;
    size_t t = i / HID;
    size_t h = i - t * HID;
    float4 a = *(const float4*)(ybuf + (t * 2 + 0) * HID + h);
    float4 b = *(const float4*)(ybuf + (t * 2 + 1) * HID + h);
    float4* o = (float4*)(out + t * HID + h);
    float4 v = *o;
    v.x += a.x + b.x; v.y += a.y + b.y; v.z += a.z + b.z; v.w += a.w + b.w;
    *o = v;
}

// ---------------------------------------------------------------------------
extern "C" void kernel_launch(void* const* d_in, const int* in_sizes, int n_in,
                              void* d_out, int out_size, void* d_ws, size_t ws_size,
                              hipStream_t stream)
{
    const float* x   = (const float*)d_in[0];
    const float* gw  = (const float*)d_in[1];
    const float* cw  = (const float*)d_in[2];
    const float* esc = (const float*)d_in[3];
    const float* ebi = (const float*)d_in[4];
    const float* wg  = (const float*)d_in[5];
    const float* wu  = (const float*)d_in[6];
    const float* wd  = (const float*)d_in[7];
    const float* sg  = (const float*)d_in[8];
    const float* su  = (const float*)d_in[9];
    const float* sd  = (const float*)d_in[10];
    float* out = (float*)d_out;

    const int T = in_sizes[0] / HID;            // 2048 tokens

    // Workspace layout
    char*  ws        = (char*)d_ws;
    int*   cnt       = (int*)ws;                                   // E counters
    size_t listBytes = (size_t)NEXP * T * sizeof(int);
    int*   listIdx   = (int*)  (ws + 256);
    float* listW     = (float*)(ws + 256 + listBytes);
    float* ybuf      = (float*)(ws + 256 + 2 * listBytes);         // [T*2][HID]

    (void)hipMemsetAsync(cnt, 0, NEXP * sizeof(int), stream);

    // 1) Router (one wave32 per token; 8 waves / 256-thread block)
    router_kernel<<<(T + 7) / 8, 256, 0, stream>>>(
        x, gw, cw, esc, ebi, cnt, listIdx, listW, T);

    const int smem = LDS_FLOATS * (int)sizeof(float);   // 311680 B <= 320KB/WGP
    (void)hipFuncSetAttribute(reinterpret_cast<const void*>(&mlp_tile_kernel<false>),
                              hipFuncAttributeMaxDynamicSharedMemorySize, smem);
    (void)hipFuncSetAttribute(reinterpret_cast<const void*>(&mlp_tile_kernel<true>),
                              hipFuncAttributeMaxDynamicSharedMemorySize, smem);

    // 2) Routed experts: grid.y = expert so same-expert tiles co-schedule and
    //    share the expert's 34.6MB weight set in the 192MB L2. Worst-case
    //    tiles per expert = T/16; surplus blocks exit after reading cnt.
    const int tiles = (T + 15) / 16;
    mlp_tile_kernel<false><<<dim3(tiles, NEXP), BLOCK, smem, stream>>>(
        x, wg, wu, wd, cnt, listIdx, listW, ybuf, T, INTER);

    // 3) Shared expert -> writes d_out directly (every element)
    mlp_tile_kernel<true><<<dim3(tiles, 1), BLOCK, smem, stream>>>(
        x, sg, su, sd, nullptr, nullptr, nullptr, out, T, SINTER);

    // 4) out += slot0 + slot1
    const size_t n = (size_t)T * HID;
    combine_kernel<<<(unsigned)((n / 4 + 255) / 256), 256, 0, stream>>>(out, ybuf, n);
}